// GCNConvolution_55662776156458
// MI455X (gfx1250) — compile-verified
//
#include <hip/hip_runtime.h>
#include <hip/hip_bf16.h>

typedef __attribute__((ext_vector_type(2))) float v2f;
typedef __attribute__((ext_vector_type(8))) float v8f;

#define FDIM 64

// ---------------- degree / normalization ----------------
__global__ void k_deg_init(float* dinv, int n) {
  int i = blockIdx.x * blockDim.x + threadIdx.x;
  if (i < n) dinv[i] = 1.0f;                      // self-loop counts once
}

__global__ void k_deg_count(const int* __restrict__ dst, float* dinv, int E) {
  int e = blockIdx.x * blockDim.x + threadIdx.x;
  if (e < E) atomicAdd(&dinv[dst[e]], 1.0f);
}

__global__ void k_deg_fin(float* dinv, int n) {
  int i = blockIdx.x * blockDim.x + threadIdx.x;
  if (i < n) dinv[i] = rsqrtf(dinv[i]);
}

__global__ void k_zero(float* p, int n) {
  int i = blockIdx.x * blockDim.x + threadIdx.x;
  if (i < n) p[i] = 0.0f;
}

// ---- fp32 WMMA GEMM: C[M x NCOLS] = A[M x 64] @ B[64 x NCOLS] (+bias) ----
// One wave per 16-row strip (NCOLS/16 accumulators). B staged in LDS once per
// block via GLOBAL_LOAD_ASYNC_TO_LDS_B128 (ASYNCcnt), then s_wait_asynccnt +
// barrier. Guards are wave-uniform so EXEC is all-ones at every WMMA.
template <int NCOLS>
__global__ void k_gemm_wmma(const float* __restrict__ A, const float* __restrict__ B,
                            const float* __restrict__ bias, float* __restrict__ C,
                            int rowTiles) {
  constexpr int CT = NCOLS / 16;
  __shared__ __align__(16) float Bs[64 * NCOLS];

  // Cooperative async copy: 64*NCOLS floats, float4 per lane per iteration.
  // 64*NCOLS is divisible by blockDim.x*4 (256*4=1024) for NCOLS in {16,64},
  // so every lane runs the same trip count (EXEC uniform).
  for (int i = threadIdx.x * 4; i < 64 * NCOLS; i += blockDim.x * 4) {
    unsigned ldsOff = (unsigned)(uintptr_t)(&Bs[i]);  // addr[31:0] = LDS byte offset
    const float* gp = B + i;
    asm volatile("global_load_async_to_lds_b128 %0, %1, off"
                 :: "v"(ldsOff), "v"(gp) : "memory");
  }
  asm volatile("s_wait_asynccnt 0x0" ::: "memory");
  __syncthreads();

  int wave = threadIdx.x >> 5;
  int rowTile = blockIdx.x * (blockDim.x >> 5) + wave;
  if (rowTile >= rowTiles) return;

  int lane = threadIdx.x & 31;
  int half = lane >> 4;        // 0: lanes 0-15 (K=k0,k0+1), 1: lanes 16-31 (K=k0+2,k0+3)
  int l16  = lane & 15;
  const float* Ap = A + (size_t)(rowTile * 16 + l16) * FDIM;

  v8f acc[CT] = {};
#pragma unroll
  for (int k0 = 0; k0 < FDIM; k0 += 4) {
    int kk = k0 + 2 * half;
    v2f a = *(const v2f*)(Ap + kk);                 // 8B-aligned contiguous pair
#pragma unroll
    for (int t = 0; t < CT; ++t) {
      v2f b;
      b.x = Bs[kk * NCOLS + t * 16 + l16];
      b.y = Bs[(kk + 1) * NCOLS + t * 16 + l16];
      acc[t] = __builtin_amdgcn_wmma_f32_16x16x4_f32(
          /*neg_a=*/false, a, /*neg_b=*/false, b,
          /*c_mod=*/(short)0, acc[t], /*reuse_a=*/false, /*reuse_b=*/false);
    }
  }

#pragma unroll
  for (int t = 0; t < CT; ++t) {
    int col = t * 16 + l16;
    float bv = bias ? bias[col] : 0.0f;
#pragma unroll
    for (int r = 0; r < 8; ++r) {
      int orow = rowTile * 16 + r + 8 * half;       // C/D: VGPR r -> M=r (lo), M=8+r (hi)
      C[(size_t)orow * NCOLS + col] = acc[t][r] + bv;
    }
  }
}

// ---------------- edge scatter: agg[dst] += h[src] * dinv[src]*dinv[dst] ----------------
// One thread per (edge, feature); features coalesced within each 64-thread group.
// Working set (~103 MB) fits the 192 MB L2, so these atomics resolve in L2.
__global__ void k_scatter(const float* __restrict__ h, const int* __restrict__ src,
                          const int* __restrict__ dst, const float* __restrict__ dinv,
                          float* __restrict__ agg, int E) {
  unsigned gid = blockIdx.x * blockDim.x + threadIdx.x;
  int e = (int)(gid >> 6);
  int f = (int)(gid & 63u);
  if (e < E) {
    int s = src[e], d = dst[e];
    float c = dinv[s] * dinv[d];
    atomicAdd(&agg[(size_t)d * FDIM + f], h[(size_t)s * FDIM + f] * c);
  }
}

// ---------------- finalize: out = relu(agg + h*dinv^2 + bias) (+resid) ----------------
__global__ void k_finalize(const float* __restrict__ agg, const float* __restrict__ h,
                           const float* __restrict__ dinv, const float* __restrict__ bias,
                           const float* __restrict__ resid, float* __restrict__ out, int n) {
  int gid = blockIdx.x * blockDim.x + threadIdx.x;
  if (gid < n) {
    int i = gid >> 6;
    int f = gid & 63;
    float di = dinv[i];
    float v = agg[gid] + h[gid] * di * di + bias[f];
    v = fmaxf(v, 0.0f);
    if (resid) v += resid[gid];
    out[gid] = v;
  }
}

// ---------------- log_softmax over 16 classes, in-place; 2 rows per wave ----------------
__global__ void k_logsoftmax16(float* __restrict__ logits, int nrows) {
  int wave = threadIdx.x >> 5;
  int lane = threadIdx.x & 31;
  int half = lane >> 4;
  int l16  = lane & 15;
  int row = blockIdx.x * ((blockDim.x >> 5) * 2) + wave * 2 + half;
  if (row >= nrows) return;
  float x = logits[(size_t)row * 16 + l16];
  float mx = x;
#pragma unroll
  for (int m = 1; m < 16; m <<= 1) mx = fmaxf(mx, __shfl_xor(mx, m, 16));
  float ex = __expf(x - mx);
  float s = ex;
#pragma unroll
  for (int m = 1; m < 16; m <<= 1) s += __shfl_xor(s, m, 16);
  logits[(size_t)row * 16 + l16] = x - mx - __logf(s);
}

extern "C" void kernel_launch(void* const* d_in, const int* in_sizes, int n_in,
                              void* d_out, int out_size, void* d_ws, size_t ws_size,
                              hipStream_t stream) {
  const float* x   = (const float*)d_in[0];
  const int*   ei  = (const int*)d_in[1];
  const float* W1  = (const float*)d_in[2];
  const float* b1  = (const float*)d_in[3];
  const float* W2  = (const float*)d_in[4];
  const float* b2  = (const float*)d_in[5];
  const float* Wl  = (const float*)d_in[6];
  const float* bl  = (const float*)d_in[7];
  float* out = (float*)d_out;

  int N = in_sizes[0] / FDIM;   // 100000
  int E = in_sizes[1] / 2;      // 800000
  const int* src = ei;          // edge_index row 0
  const int* dst = ei + E;      // edge_index row 1

  // workspace layout (floats): dinv[N] | h[N*64] | agg[N*64] | h1[N*64]  (~77 MB)
  float* ws   = (float*)d_ws;
  float* dinv = ws;
  float* h    = dinv + N;
  float* agg  = h + (size_t)N * FDIM;
  float* h1   = agg + (size_t)N * FDIM;

  const int B = 256;
  int nf = N * FDIM;
  unsigned scatterBlocks = (unsigned)(((unsigned long long)E * FDIM + B - 1) / B);
  int rowTiles = N / 16;
  int gemmBlocks = (rowTiles + 7) / 8;   // 8 waves (strips) per block

  // normalization coefficients (shared by both layers)
  k_deg_init <<<(N + B - 1) / B, B, 0, stream>>>(dinv, N);
  k_deg_count<<<(E + B - 1) / B, B, 0, stream>>>(dst, dinv, E);
  k_deg_fin  <<<(N + B - 1) / B, B, 0, stream>>>(dinv, N);

  // ---- layer 1 ----
  k_gemm_wmma<64><<<gemmBlocks, B, 0, stream>>>(x, W1, nullptr, h, rowTiles);
  k_zero     <<<(nf + B - 1) / B, B, 0, stream>>>(agg, nf);
  k_scatter  <<<scatterBlocks, B, 0, stream>>>(h, src, dst, dinv, agg, E);
  k_finalize <<<(nf + B - 1) / B, B, 0, stream>>>(agg, h, dinv, b1, nullptr, h1, nf);

  // ---- layer 2 (+ residual), finalize in-place into agg ----
  k_gemm_wmma<64><<<gemmBlocks, B, 0, stream>>>(h1, W2, nullptr, h, rowTiles);
  k_zero     <<<(nf + B - 1) / B, B, 0, stream>>>(agg, nf);
  k_scatter  <<<scatterBlocks, B, 0, stream>>>(h, src, dst, dinv, agg, E);
  k_finalize <<<(nf + B - 1) / B, B, 0, stream>>>(agg, h, dinv, b2, h1, agg, nf);

  // ---- classifier head: logits -> d_out, then in-place log_softmax ----
  k_gemm_wmma<16><<<gemmBlocks, B, 0, stream>>>(agg, Wl, bl, out, rowTiles);
  k_logsoftmax16<<<(N + 15) / 16, B, 0, stream>>>(out, N);
}